// Correction_Module_dense_44667659878981
// MI455X (gfx1250) — compile-verified
//
#include <hip/hip_runtime.h>

#define TILE      2048
#define THREADS   256
#define PER_THREAD (TILE / THREADS)   // 8 floats = 2 x float4 per thread
#define KSIG      4.0f

__device__ __forceinline__ float cleanf(float v) {
    // nan_checker: NaN/Inf -> 0
    return __builtin_isfinite(v) ? v : 0.0f;
}

// --- Pass 1: hoist sqrt out of the 64M-element stream (only 16K sqrts) ----
__global__ void __launch_bounds__(THREADS)
bounds_kernel(const float* __restrict__ mean_grad,
              const float* __restrict__ var_grad,
              float* __restrict__ lower,
              float* __restrict__ upper,
              int N)
{
    int i = blockIdx.x * blockDim.x + threadIdx.x;
    if (i < N) {
        float s = KSIG * sqrtf(var_grad[i]);
        float m = mean_grad[i];
        lower[i] = m - s;
        upper[i] = m + s;
    }
}

// --- Pass 2: streaming kernel; tile staged to LDS via CDNA5 async DMA -----
// a/b = lower/upper if haveBounds, else mean/var (sqrt done inline).
__global__ void __launch_bounds__(THREADS)
corr_kernel(const float* __restrict__ x,
            const float* __restrict__ a,
            const float* __restrict__ b,
            float* __restrict__ out,
            int N, int tilesPerRow, int haveBounds)
{
    __shared__ __align__(16) float tile[TILE];
    __shared__ float prev_raw;

    const int tid      = threadIdx.x;
    const int tileIdx  = blockIdx.x;
    const int row      = tileIdx / tilesPerRow;
    const int tcol     = (tileIdx % tilesPerRow) * TILE;
    const size_t rowBase = (size_t)row * (size_t)N;

    const int li0 = tid * PER_THREAD;                 // local float index base
    const float* g0 = x + rowBase + tcol + li0;       // 16B aligned
    unsigned l0 = (unsigned)(uintptr_t)(&tile[li0]);  // LDS byte address (low 32 bits)

    // Async copy: INST_OFFSET applies to both global and LDS addresses,
    // so one address pair drives both 16B chunks of this thread.
    asm volatile("global_load_async_to_lds_b128 %0, %1, off\n\t"
                 "global_load_async_to_lds_b128 %0, %1, off offset:16"
                 :: "v"(l0), "v"(g0)
                 : "memory");

    if (tid == 0) {
        int pcol = (tcol == 0) ? (N - 1) : (tcol - 1);  // circular predecessor
        prev_raw = x[rowBase + pcol];
    }

    asm volatile("s_wait_asynccnt 0" ::: "memory");
    __syncthreads();

#pragma unroll
    for (int c = 0; c < PER_THREAD / 4; ++c) {
        const int li  = li0 + 4 * c;
        const int col = tcol + li;

        float4 xv = *reinterpret_cast<const float4*>(&tile[li]);
        float xp_raw = (li == 0) ? prev_raw : tile[li - 1];

        float x0 = cleanf(xv.x), x1 = cleanf(xv.y);
        float x2 = cleanf(xv.z), x3 = cleanf(xv.w);
        float xp = cleanf(xp_raw);

        float gq0 = x0 - xp, gq1 = x1 - x0, gq2 = x2 - x1, gq3 = x3 - x2;

        float4 av = *reinterpret_cast<const float4*>(a + col);
        float4 bv = *reinterpret_cast<const float4*>(b + col);

        float lo0, lo1, lo2, lo3, hi0, hi1, hi2, hi3;
        if (haveBounds) {
            lo0 = av.x; lo1 = av.y; lo2 = av.z; lo3 = av.w;
            hi0 = bv.x; hi1 = bv.y; hi2 = bv.z; hi3 = bv.w;
        } else {
            float s0 = KSIG * sqrtf(bv.x), s1 = KSIG * sqrtf(bv.y);
            float s2 = KSIG * sqrtf(bv.z), s3 = KSIG * sqrtf(bv.w);
            lo0 = av.x - s0; hi0 = av.x + s0;
            lo1 = av.y - s1; hi1 = av.y + s1;
            lo2 = av.z - s2; hi2 = av.z + s2;
            lo3 = av.w - s3; hi3 = av.w + s3;
        }

        float4 o;
        o.x = (gq0 < lo0 || gq0 > hi0) ? 0.0f : x0;
        o.y = (gq1 < lo1 || gq1 > hi1) ? 0.0f : x1;
        o.z = (gq2 < lo2 || gq2 > hi2) ? 0.0f : x2;
        o.w = (gq3 < lo3 || gq3 > hi3) ? 0.0f : x3;

        *reinterpret_cast<float4*>(out + rowBase + col) = o;
    }
}

// --- Generic scalar fallback (only if N is not a multiple of TILE) --------
__global__ void corr_scalar(const float* __restrict__ x,
                            const float* __restrict__ a,
                            const float* __restrict__ b,
                            float* __restrict__ out,
                            int N, long long total, int haveBounds)
{
    long long i = (long long)blockIdx.x * blockDim.x + threadIdx.x;
    if (i >= total) return;
    int col = (int)(i % (long long)N);
    long long rowBase = i - col;
    float xi = cleanf(x[i]);
    float xp = cleanf(x[rowBase + ((col == 0) ? N - 1 : col - 1)]);
    float g = xi - xp;
    float lo, hi;
    if (haveBounds) { lo = a[col]; hi = b[col]; }
    else { float s = KSIG * sqrtf(b[col]); lo = a[col] - s; hi = a[col] + s; }
    out[i] = (g < lo || g > hi) ? 0.0f : xi;
}

extern "C" void kernel_launch(void* const* d_in, const int* in_sizes, int n_in,
                              void* d_out, int out_size, void* d_ws, size_t ws_size,
                              hipStream_t stream)
{
    const float* x         = (const float*)d_in[0];
    const float* mean_grad = (const float*)d_in[1];
    const float* var_grad  = (const float*)d_in[2];
    float* out = (float*)d_out;

    const int N = in_sizes[1];                 // 16384
    const long long total = in_sizes[0];       // B * N
    const int Bn = (int)(total / (long long)N);

    const bool haveBounds = (ws_size >= (size_t)2 * (size_t)N * sizeof(float));
    const float* a = mean_grad;
    const float* b = var_grad;
    if (haveBounds) {
        float* lower = (float*)d_ws;
        float* upper = lower + N;
        bounds_kernel<<<(N + THREADS - 1) / THREADS, THREADS, 0, stream>>>(
            mean_grad, var_grad, lower, upper, N);
        a = lower;
        b = upper;
    }

    if (N % TILE == 0) {
        const int tilesPerRow = N / TILE;      // 8
        const int nTiles = Bn * tilesPerRow;   // 32768 blocks
        corr_kernel<<<nTiles, THREADS, 0, stream>>>(
            x, a, b, out, N, tilesPerRow, haveBounds ? 1 : 0);
    } else {
        const long long nb = (total + THREADS - 1) / THREADS;
        corr_scalar<<<(int)nb, THREADS, 0, stream>>>(
            x, a, b, out, N, total, haveBounds ? 1 : 0);
    }
}